// GCF_21706764714013
// MI455X (gfx1250) — compile-verified
//
#include <hip/hip_runtime.h>
#include <hip/hip_bf16.h>

typedef __attribute__((ext_vector_type(2))) float v2f;
typedef __attribute__((ext_vector_type(8))) float v8f;

#define USER_NUM 60000
#define ITEM_NUM 40000
#define NTOT     100000
#define DDIM     64
#define BATCH    16384

// ---------------------------------------------------------------------------
// 0) zero workspace (4 aggregate matrices, N x 64 each)
// ---------------------------------------------------------------------------
__global__ void zero_ws_kernel(float* __restrict__ ws, long n) {
    long i = (long)blockIdx.x * blockDim.x + threadIdx.x;
    long stride = (long)gridDim.x * blockDim.x;
    for (; i < n; i += stride) ws[i] = 0.0f;
}

// ---------------------------------------------------------------------------
// 1) Dual SpMM: one pass over a COO matrix computes  agg_f += v * x  and
//    agg_i += v * x*x  (x gathered once, squared in registers).
//    One wave (32 lanes) per nnz edge; each lane owns a float2 column pair.
// ---------------------------------------------------------------------------
__global__ void spmm_dual_kernel(const int* __restrict__ rows,
                                 const int* __restrict__ cols,
                                 const float* __restrict__ vals,
                                 const float* __restrict__ uEmbd,
                                 const float* __restrict__ iEmbd,
                                 float* __restrict__ aggF,
                                 float* __restrict__ aggI,
                                 int nnz) {
    int tid = blockIdx.x * blockDim.x + threadIdx.x;
    int e = tid >> 5;              // wave-uniform edge id
    if (e >= nnz) return;
    int p = (tid & 31) * 2;        // column pair base (0,2,...,62)

    int r = rows[e];
    int c = cols[e];
    float v = vals[e];

    const float* src = (c < USER_NUM) ? (uEmbd + (long)c * DDIM)
                                      : (iEmbd + (long)(c - USER_NUM) * DDIM);
    v2f x = *(const v2f*)(src + p);

    float fx = v * x[0];
    float fy = v * x[1];
    float ix = fx * x[0];          // v * x^2
    float iy = fy * x[1];

    float* dF = aggF + (long)r * DDIM + p;
    float* dI = aggI + (long)r * DDIM + p;
    __hip_atomic_fetch_add(dF + 0, fx, __ATOMIC_RELAXED, __HIP_MEMORY_SCOPE_AGENT);
    __hip_atomic_fetch_add(dF + 1, fy, __ATOMIC_RELAXED, __HIP_MEMORY_SCOPE_AGENT);
    __hip_atomic_fetch_add(dI + 0, ix, __ATOMIC_RELAXED, __HIP_MEMORY_SCOPE_AGENT);
    __hip_atomic_fetch_add(dI + 1, iy, __ATOMIC_RELAXED, __HIP_MEMORY_SCOPE_AGENT);
}

// ---------------------------------------------------------------------------
// 2) out = relu( sum_m  agg_m @ W_m  + sum(b) ), written to finalEmbd[:,64:128];
//    finalEmbd[:,0:64] = original features.  fp32 WMMA 16x16x4.
//    One wave computes a 16-row x 64-col tile (4 accumulators).
// ---------------------------------------------------------------------------
__global__ void gnn_gemm_kernel(const float* __restrict__ A0, const float* __restrict__ A1,
                                const float* __restrict__ A2, const float* __restrict__ A3,
                                const float* __restrict__ W0, const float* __restrict__ W1,
                                const float* __restrict__ W2, const float* __restrict__ W3,
                                const float* __restrict__ b0, const float* __restrict__ b1,
                                const float* __restrict__ b2, const float* __restrict__ b3,
                                const float* __restrict__ uEmbd, const float* __restrict__ iEmbd,
                                float* __restrict__ finalEmbd) {
    int wave = (blockIdx.x * blockDim.x + threadIdx.x) >> 5;
    int lane = threadIdx.x & 31;
    const int mtiles = NTOT / 16;          // 6250, exact
    if (wave >= mtiles) return;            // wave-uniform exit -> EXEC full for WMMA
    int rowbase = wave * 16;
    int m = lane & 15;
    int khalf = lane >> 4;

    v8f acc[4];
    #pragma unroll
    for (int nt = 0; nt < 4; ++nt) {
        int col = nt * 16 + m;
        float bs = b0[col] + b1[col] + b2[col] + b3[col];
        #pragma unroll
        for (int v = 0; v < 8; ++v) acc[nt][v] = bs;
    }

    const float* As[4] = {A0, A1, A2, A3};
    const float* Ws[4] = {W0, W1, W2, W3};
    for (int mm = 0; mm < 4; ++mm) {
        const float* A = As[mm];
        const float* W = Ws[mm];
        #pragma unroll
        for (int k4 = 0; k4 < DDIM; k4 += 4) {
            int k0 = k4 + 2 * khalf;
            // A 16x4 fp32 layout: lane m/khalf holds row m, K = k0, k0+1
            v2f a = *(const v2f*)(A + (long)(rowbase + m) * DDIM + k0);
            #pragma unroll
            for (int nt = 0; nt < 4; ++nt) {
                int col = nt * 16 + m;
                v2f b;
                b[0] = W[k0 * DDIM + col];
                b[1] = W[(k0 + 1) * DDIM + col];
                acc[nt] = __builtin_amdgcn_wmma_f32_16x16x4_f32(
                    false, a, false, b, (short)0, acc[nt], false, false);
            }
        }
    }

    // epilogue: concat(original features, relu(out)) into finalEmbd [N,128]
    #pragma unroll
    for (int nt = 0; nt < 4; ++nt) {
        int col = nt * 16 + m;
        #pragma unroll
        for (int v = 0; v < 8; ++v) {
            int row = rowbase + v + 8 * khalf;
            float fv = (row < USER_NUM) ? uEmbd[(long)row * DDIM + col]
                                        : iEmbd[(long)(row - USER_NUM) * DDIM + col];
            finalEmbd[(long)row * 128 + col]      = fv;
            finalEmbd[(long)row * 128 + 64 + col] = fmaxf(acc[nt][v], 0.0f);
        }
    }
}

// ---------------------------------------------------------------------------
// 3) Batch gather + prediction MLP.  One wave per 16 batch rows.
//    Layer 1 (256->64) via fp32 WMMA from LDS; layers 2/3 via VALU + shuffle.
// ---------------------------------------------------------------------------
__global__ __launch_bounds__(32) void predict_kernel(
        const int* __restrict__ userIdx, const int* __restrict__ itemIdx,
        const float* __restrict__ finalEmbd,
        const float* __restrict__ Wt1, const float* __restrict__ bt1,
        const float* __restrict__ Wt2, const float* __restrict__ bt2,
        const float* __restrict__ Wt3, const float* __restrict__ bt3,
        float* __restrict__ pred, float* __restrict__ outU, float* __restrict__ outI) {
    __shared__ __align__(16) float embd[16][256];
    __shared__ __align__(16) float h1[16][64];
    int lane = threadIdx.x;
    int base = blockIdx.x * 16;

    // gather 16 (user,item) rows; also emit userEmbd/itemEmbd outputs
    for (int r = 0; r < 16; ++r) {
        int b = base + r;
        const float* up = finalEmbd + (long)userIdx[b] * 128;
        const float* ip = finalEmbd + (long)(itemIdx[b] + USER_NUM) * 128;
        float4 u  = *(const float4*)(up + lane * 4);
        float4 it = *(const float4*)(ip + lane * 4);
        *(float4*)&embd[r][lane * 4]       = u;
        *(float4*)&embd[r][128 + lane * 4] = it;
        *(float4*)(outU + (long)b * 128 + lane * 4) = u;
        *(float4*)(outI + (long)b * 128 + lane * 4) = it;
    }
    __syncthreads();

    int m = lane & 15;
    int khalf = lane >> 4;
    v8f acc[4];
    #pragma unroll
    for (int nt = 0; nt < 4; ++nt) {
        float bs = bt1[nt * 16 + m];
        #pragma unroll
        for (int v = 0; v < 8; ++v) acc[nt][v] = bs;
    }
    for (int k4 = 0; k4 < 256; k4 += 4) {
        int k0 = k4 + 2 * khalf;
        v2f a = *(const v2f*)&embd[m][k0];
        #pragma unroll
        for (int nt = 0; nt < 4; ++nt) {
            int col = nt * 16 + m;
            v2f b;
            b[0] = Wt1[k0 * 64 + col];
            b[1] = Wt1[(k0 + 1) * 64 + col];
            acc[nt] = __builtin_amdgcn_wmma_f32_16x16x4_f32(
                false, a, false, b, (short)0, acc[nt], false, false);
        }
    }
    #pragma unroll
    for (int nt = 0; nt < 4; ++nt) {
        int col = nt * 16 + m;
        #pragma unroll
        for (int v = 0; v < 8; ++v)
            h1[v + 8 * khalf][col] = fmaxf(acc[nt][v], 0.0f);
    }
    __syncthreads();

    // layer 2 (64->32): lane owns one h2 column; layer 3 (32->1): shuffle reduce
    float w3 = Wt3[lane];
    float bias2 = bt2[lane];
    float bias3 = bt3[0];
    for (int r = 0; r < 16; ++r) {
        float h2 = bias2;
        #pragma unroll 8
        for (int k = 0; k < 64; ++k) h2 += h1[r][k] * Wt2[k * 32 + lane];
        float p = h2 * w3;
        for (int off = 16; off > 0; off >>= 1) p += __shfl_xor(p, off, 32);
        if (lane == 0) pred[base + r] = p + bias3;
    }
}

// ---------------------------------------------------------------------------
extern "C" void kernel_launch(void* const* d_in, const int* in_sizes, int n_in,
                              void* d_out, int out_size, void* d_ws, size_t ws_size,
                              hipStream_t stream) {
    const int*   userIdx  = (const int*)d_in[0];
    const int*   itemIdx  = (const int*)d_in[1];
    const int*   L_rows   = (const int*)d_in[2];
    const int*   L_cols   = (const int*)d_in[3];
    const float* L_vals   = (const float*)d_in[4];
    const int*   Lui_rows = (const int*)d_in[5];
    const int*   Lui_cols = (const int*)d_in[6];
    const float* Lui_vals = (const float*)d_in[7];
    const float* uEmbd    = (const float*)d_in[8];
    const float* iEmbd    = (const float*)d_in[9];
    const float* W_lin  = (const float*)d_in[10]; const float* b_lin  = (const float*)d_in[11];
    const float* W_lin1 = (const float*)d_in[12]; const float* b_lin1 = (const float*)d_in[13];
    const float* W_ia   = (const float*)d_in[14]; const float* b_ia   = (const float*)d_in[15];
    const float* W_ia1  = (const float*)d_in[16]; const float* b_ia1  = (const float*)d_in[17];
    const float* W_t1 = (const float*)d_in[18]; const float* b_t1 = (const float*)d_in[19];
    const float* W_t2 = (const float*)d_in[20]; const float* b_t2 = (const float*)d_in[21];
    const float* W_t3 = (const float*)d_in[22]; const float* b_t3 = (const float*)d_in[23];

    int nnzL = in_sizes[2];
    int nnzU = in_sizes[5];

    // d_out layout: pred[B] | userEmbd[B*128] | itemEmbd[B*128] | finalEmbd[N*128]
    float* out  = (float*)d_out;
    float* pred = out;
    float* outU = out + BATCH;
    float* outI = outU + (long)BATCH * 128;
    float* finalEmbd = outI + (long)BATCH * 128;

    const long N64 = (long)NTOT * DDIM;
    float* ws    = (float*)d_ws;
    float* aggLf = ws;
    float* aggLi = ws + N64;
    float* aggUf = ws + 2 * N64;
    float* aggUi = ws + 3 * N64;

    zero_ws_kernel<<<4096, 256, 0, stream>>>(ws, 4 * N64);

    {
        long threads = (long)nnzL * 32;
        int blocks = (int)((threads + 255) / 256);
        spmm_dual_kernel<<<blocks, 256, 0, stream>>>(L_rows, L_cols, L_vals,
                                                     uEmbd, iEmbd, aggLf, aggLi, nnzL);
    }
    {
        long threads = (long)nnzU * 32;
        int blocks = (int)((threads + 255) / 256);
        spmm_dual_kernel<<<blocks, 256, 0, stream>>>(Lui_rows, Lui_cols, Lui_vals,
                                                     uEmbd, iEmbd, aggUf, aggUi, nnzU);
    }
    {
        int waves  = NTOT / 16;                 // 6250
        int blocks = (waves * 32 + 255) / 256;  // 8 waves per block
        gnn_gemm_kernel<<<blocks, 256, 0, stream>>>(aggLf, aggUf, aggLi, aggUi,
                                                    W_lin, W_lin1, W_ia, W_ia1,
                                                    b_lin, b_lin1, b_ia, b_ia1,
                                                    uEmbd, iEmbd, finalEmbd);
    }
    predict_kernel<<<BATCH / 16, 32, 0, stream>>>(userIdx, itemIdx, finalEmbd,
                                                  W_t1, b_t1, W_t2, b_t2, W_t3, b_t3,
                                                  pred, outU, outI);
}